// KANLayer_11089605558325
// MI455X (gfx1250) — compile-verified
//
#include <hip/hip_runtime.h>
#include <hip/hip_bf16.h>

// KAN spline layer: out[b, j, q*1024 + r] = poly(x[b, r*64 + j, q], coef[j])
// B=8, N=65536, D=64, G=5. Pure bandwidth problem (268 MB @ 23.3 TB/s ~= 11.5us).
// Strategy: TDM (tensor_load_to_lds) gathers the strided 128x64 f32 tile
// (rows of 256B, 16KB apart) into LDS with hardware row padding (pitch 65
// dwords -> conflict-free transposed reads); compute spline in registers;
// write 512B contiguous non-temporal bursts per wave (streaming, no reuse).

typedef __attribute__((ext_vector_type(4))) unsigned int v4u;
typedef __attribute__((ext_vector_type(8))) int          v8i;
typedef __attribute__((ext_vector_type(4))) int          v4i;
typedef __attribute__((ext_vector_type(4))) float        v4f;

#define TR      128                 // r rows per tile
#define TQ      64                  // q per tile (contiguous in input)
#define PITCH   65                  // LDS dwords per row (64 data + 1 TDM pad)
#define TILE_F  (TR * PITCH)        // 8320 floats = 33280 B (16B aligned)
#define COEF_OFF TILE_F

__global__ __launch_bounds__(256) void kan_tdm_kernel(const float* __restrict__ x,
                                                      const float* __restrict__ coef,
                                                      float* __restrict__ out) {
    // Single static shared allocation -> LDS offset 0 for the TDM descriptor.
    __shared__ __align__(16) float lds[TILE_F + 32];

    const int blk  = (int)blockIdx.x;
    const int rblk = blk & 7;          // 8 r-blocks of 128 rows
    const int j    = (blk >> 3) & 63;  // coefficient row / output channel
    const int b    = blk >> 9;         // batch

    const long long ND = 65536LL * 64LL;
    const float* __restrict__ src = x   + (long long)b * ND + (long long)rblk * TR * 4096 + j * 64;
    float*       __restrict__ dst = out + (long long)b * ND + (long long)j * 65536 + rblk * TR;

    // Stage this channel's 20 spline coefficients (5 segs x 4) into LDS.
    if (threadIdx.x < 20) lds[COEF_OFF + threadIdx.x] = coef[j * 20 + threadIdx.x];

    // ---- TDM: strided 128x64 f32 tile -> LDS, padded to pitch 65 dwords ----
    const int wave = __builtin_amdgcn_readfirstlane((int)(threadIdx.x >> 5));
    if (wave == 0) {   // wave-uniform scalar branch: exactly one TDM issue per block
#if __has_builtin(__builtin_amdgcn_tensor_load_to_lds)
        unsigned long long ga = (unsigned long long)(uintptr_t)src;
        // D# group 0: count=1 valid descriptor, lds_addr=0, 57b global addr, type=2
        v4u g0 = { 1u,
                   0u,
                   (unsigned)(ga & 0xFFFFFFFFu),
                   (unsigned)((ga >> 32) & 0x1FFFFFFu) | (2u << 30) };
        // D# group 1: data_size=4B(code 2), pad_enable, pad_interval=64dw(code 5),
        // pad_amount=1dw(code 0); tensor_dim0=64, tensor_dim1=128;
        // tile_dim0=64, tile_dim1=128; tensor_dim0_stride=4096 elements.
        v8i g1 = { (int)((2u << 16) | (1u << 20) | (5u << 22)),
                   (int)(64u  << 16),   // tensor_dim0 lo16 (hi16 of atomic addr = 0)
                   (int)(128u << 16),   // tensor_dim0 hi16=0 | tensor_dim1 lo16
                   (int)(64u  << 16),   // tensor_dim1 hi16=0 | tile_dim0
                   (int)128,            // tile_dim1 (tile_dim2=0)
                   (int)4096,           // tensor_dim0_stride lo32
                   0, 0 };
        v4i gz = { 0, 0, 0, 0 };             // groups 2/3 unused (2D tensor)
        v8i z8 = { 0, 0, 0, 0, 0, 0, 0, 0 }; // extra operand (clang-23 6-arg form)
        __builtin_amdgcn_tensor_load_to_lds(g0, g1, gz, gz, z8, 0);
        __builtin_amdgcn_s_wait_tensorcnt(0);
#else
        for (int e = (int)threadIdx.x; e < TR * TQ; e += 32) {
            int r = e >> 6, q = e & 63;
            lds[r * PITCH + q] = src[(long long)r * 4096 + q];
        }
#endif
    }
    __syncthreads();

    const v4f* __restrict__ cf = (const v4f*)&lds[COEF_OFF];

    // 2048 output float4s per tile; lane l of a wave handles r = 4l..4l+3 of one
    // q column -> 512B contiguous global_store_b128 (non-temporal) per wave.
#pragma unroll
    for (int i = 0; i < 8; ++i) {
        const int e  = (int)threadIdx.x + i * 256;  // float4 index in tile
        const int q  = e >> 5;                      // TR/4 = 32 float4 per column
        const int rv = (e & 31) << 2;

        v4f o;
#pragma unroll
        for (int k = 0; k < 4; ++k) {
            float xv  = lds[(rv + k) * PITCH + q];
            float xc  = fminf(fmaxf(xv, 0.0f), 1.0f);
            int   idx = (int)(xc * 5.0f);           // floor (xc >= 0)
            idx       = idx > 4 ? 4 : idx;
            float t   = xc - (float)idx * 0.2f;
            v4f   c   = cf[idx];                    // ds_load_b128, broadcast-friendly
            o[k]      = fmaf(fmaf(fmaf(c[3], t, c[2]), t, c[1]), t, c[0]);
        }
        __builtin_nontemporal_store(o, (v4f*)(dst + (long long)q * 1024 + rv));
    }
}

extern "C" void kernel_launch(void* const* d_in, const int* in_sizes, int n_in,
                              void* d_out, int out_size, void* d_ws, size_t ws_size,
                              hipStream_t stream) {
    (void)in_sizes; (void)n_in; (void)out_size; (void)d_ws; (void)ws_size;
    const float* x    = (const float*)d_in[0];
    const float* coef = (const float*)d_in[1];
    float* out        = (float*)d_out;
    // 8 batches * 64 channels * 8 r-blocks = 4096 blocks
    hipLaunchKernelGGL(kan_tdm_kernel, dim3(4096), dim3(256), 0, stream, x, coef, out);
}